// MinLSTM_36120674960013
// MI455X (gfx1250) — compile-verified
//
#include <hip/hip_runtime.h>

// ---------------- problem constants ----------------
#define Bb   8
#define Tt   4096
#define Dd   512
#define Hh   512
#define MTOT (Bb * Tt)        // 32768 rows

// ---------------- GEMM tiling ----------------------
#define BM   128
#define BN   64
#define BK   32
#define SXK  36               // padded LDS stride for x tile (16B aligned rows, conflict-free)
#define SWK  34               // padded LDS stride for W tiles (transposed [N][K])

// ---------------- scan chunking --------------------
#define TC      64
#define NCHUNK  (Tt / TC)     // 64

// Async Global->LDS staging of the x tile (gfx1250 GLOBAL_LOAD_ASYNC_TO_LDS_B128,
// tracked by ASYNCcnt). Flip to 0 to fall back to the synchronous path.
#define USE_ASYNC_LDS 1

typedef float v2f __attribute__((ext_vector_type(2)));
typedef float v8f __attribute__((ext_vector_type(8)));

// stable softplus: log(1 + e^x)
__device__ __forceinline__ float sp_(float x) {
    return fmaxf(x, 0.0f) + log1pf(__expf(-fabsf(x)));
}
// stable logaddexp
__device__ __forceinline__ float lae_(float a, float b) {
    float m = fmaxf(a, b);
    return m + log1pf(__expf(-fabsf(a - b)));
}

#if USE_ASYNC_LDS
// LDS byte offset of a pointer into a __shared__ array (addrspacecast to as(3)).
__device__ __forceinline__ unsigned lds_off_of(const float* p) {
    return (unsigned)(uintptr_t)(__attribute__((address_space(3))) const float*)p;
}
#endif

// =====================================================================
// Kernel 1: fused triple GEMM (x@Wf, x@Wi, x@Wh) + log-space gating.
// Emits A = log_f and V = log_i + log_g(zh), each [MTOT, H] fp32.
// wave32; per-wave 16x64 strip via V_WMMA_F32_16X16X4_F32.
// Double-buffered LDS; x tile staged via async Global->LDS DMA.
// =====================================================================
__global__ __launch_bounds__(256) void
minlstm_gemm(const float* __restrict__ x,
             const float* __restrict__ Wf, const float* __restrict__ bf,
             const float* __restrict__ Wi, const float* __restrict__ bi,
             const float* __restrict__ Wh, const float* __restrict__ bh,
             float* __restrict__ Aout, float* __restrict__ Vout)
{
    __shared__ float sX[2][BM * SXK];        // x tile  [BM][BK]  (row-major, padded)
    __shared__ float sW[2][3][BN * SWK];     // W tiles [BN][BK]  (TRANSPOSED, padded)

    const int tid  = threadIdx.x;
    const int lane = tid & 31;               // wave32
    const int wv   = tid >> 5;               // 8 waves
    const int half = lane >> 4;              // ISA lane-half split
    const int l15  = lane & 15;

    const int mblk = blockIdx.x * BM;
    const int nblk = blockIdx.y * BN;

    const float* Wm[3] = { Wf, Wi, Wh };

    const v8f zero8 = {0.f,0.f,0.f,0.f,0.f,0.f,0.f,0.f};
    v8f acc[3][4];
#pragma unroll
    for (int m = 0; m < 3; ++m)
#pragma unroll
        for (int j = 0; j < 4; ++j) acc[m][j] = zero8;

    // ---- tile stagers -------------------------------------------------
    auto stage_x = [&](int k0, int buf) {
#pragma unroll
        for (int i = 0; i < 4; ++i) {
            int q  = tid + 256 * i;                  // 0..1023
            int r  = q >> 3;
            int c4 = (q & 7) << 2;
#if USE_ASYNC_LDS
            unsigned voff   = (unsigned)((mblk + r) * Dd + k0 + c4) * 4u; // byte offset < 2^27
            unsigned ldsoff = lds_off_of(&sX[buf][r * SXK + c4]);
            // VDST = LDS byte offset, VADDR = 32-bit voffset, SADDR = 64-bit base (GVS)
            asm volatile("global_load_async_to_lds_b128 %0, %1, %2"
                         :
                         : "v"(ldsoff), "v"(voff), "s"(x)
                         : "memory");
#else
            float4 v = *(const float4*)&x[(size_t)(mblk + r) * Dd + k0 + c4];
            *(float4*)&sX[buf][r * SXK + c4] = v;    // 16B aligned: SXK=36 -> 144B rows
#endif
        }
    };
    auto stage_w = [&](int k0, int buf) {
#pragma unroll
        for (int m = 0; m < 3; ++m) {
            const float* W = Wm[m];
#pragma unroll
            for (int i = 0; i < 2; ++i) {
                int q  = tid + 256 * i;              // 0..511
                int kr = q >> 4;
                int n4 = (q & 15) << 2;
                float4 v = *(const float4*)&W[(size_t)(k0 + kr) * Hh + nblk + n4];
                sW[buf][m][(n4 + 0) * SWK + kr] = v.x;   // transpose into [N][K]
                sW[buf][m][(n4 + 1) * SWK + kr] = v.y;
                sW[buf][m][(n4 + 2) * SWK + kr] = v.z;
                sW[buf][m][(n4 + 3) * SWK + kr] = v.w;
            }
        }
    };

    // ---- prologue: stage chunk 0 into buffer 0 ----
    stage_x(0, 0);
    stage_w(0, 0);

    int buf = 0;
    for (int k0 = 0; k0 < Dd; k0 += BK) {
#if USE_ASYNC_LDS
        asm volatile("s_wait_asynccnt 0x0" ::: "memory");   // my async tile landed
#endif
        __syncthreads();                                    // everyone's tile visible

        // kick off next chunk into the other buffer while we compute
        if (k0 + BK < Dd) {
            stage_x(k0 + BK, buf ^ 1);
            stage_w(k0 + BK, buf ^ 1);
        }

        // ---- 8 x (K=4) WMMA steps, 12 WMMAs each ----
#pragma unroll
        for (int kk = 0; kk < BK; kk += 4) {
            // A frag: lane l15 = row M, VGPR pair holds K = kk + 2*half + {0,1}
            v2f afrag = *(const v2f*)&sX[buf][(16 * wv + l15) * SXK + kk + 2 * half];
#pragma unroll
            for (int m = 0; m < 3; ++m) {
#pragma unroll
                for (int j = 0; j < 4; ++j) {
                    // B frag: lane l15 = col N, contiguous K pair from transposed tile
                    v2f bfrag = *(const v2f*)&sW[buf][m][(j * 16 + l15) * SWK + kk + 2 * half];
                    acc[m][j] = __builtin_amdgcn_wmma_f32_16x16x4_f32(
                        false, afrag, false, bfrag, (short)0, acc[m][j], false, false);
                }
            }
        }
        buf ^= 1;
    }

    // ---- fused epilogue: bias + log-space gates, write A/V ----
#pragma unroll
    for (int j = 0; j < 4; ++j) {
        int n = nblk + j * 16 + l15;
        float bfv = bf[n], biv = bi[n], bhv = bh[n];
#pragma unroll
        for (int jj = 0; jj < 8; ++jj) {
            int mg = mblk + 16 * wv + jj + 8 * half;   // C layout: VGPR jj -> M=jj(+8)
            float zf = acc[0][j][jj] + bfv;
            float zi = acc[1][j][jj] + biv;
            float zh = acc[2][j][jj] + bhv;
            float diff  = sp_(-zf) - sp_(-zi);
            float log_f = -sp_(diff);
            float log_i = -sp_(-diff);
            float lg    = (zh >= 0.0f) ? __logf(zh + 0.5f) : -sp_(-zh);
            size_t idx = (size_t)mg * Hh + n;
            Aout[idx] = log_f;
            Vout[idx] = log_i + lg;
        }
    }
}

// =====================================================================
// Kernel 2a: per-(b,chunk,h) logsumexp of w_t = V - A over its TC steps.
// =====================================================================
__global__ __launch_bounds__(256) void
scan_chunk_total(const float* __restrict__ A, const float* __restrict__ V,
                 float* __restrict__ S)
{
    int g     = blockIdx.x * 256 + threadIdx.x;     // B*NCHUNK*H threads
    int h     = g & (Hh - 1);
    int t     = g >> 9;
    int chunk = t & (NCHUNK - 1);
    int b     = t >> 6;
    size_t base = ((size_t)(b * Tt + chunk * TC)) * Hh + h;

    float s = V[base] - A[base];
    for (int i = 1; i < TC; ++i) {
        size_t p = base + (size_t)i * Hh;
        s = lae_(s, V[p] - A[p]);
    }
    S[((size_t)(b * NCHUNK + chunk)) * Hh + h] = s;
}

// =====================================================================
// Kernel 2b: per-channel exclusive logaddexp prefix over chunk totals,
// seeded with log(0.5) (the t=0 element of log_val, a_star=0).
// =====================================================================
__global__ __launch_bounds__(256) void
scan_prefix(const float* __restrict__ S, float* __restrict__ P)
{
    int g = blockIdx.x * 256 + threadIdx.x;         // B*H threads
    int h = g & (Hh - 1);
    int b = g >> 9;
    float run = -0.69314718055994531f;              // log(0.5)
    for (int k = 0; k < NCHUNK; ++k) {
        size_t p = ((size_t)(b * NCHUNK + k)) * Hh + h;
        P[p] = run;
        run = lae_(run, S[p]);
    }
}

// =====================================================================
// Kernel 2c: emit out[t] = exp(a_star[t] + cumlogsumexp_t)  for t>=1,
// out[0] = exp(log 0.5) = 0.5.  out is [B, T+1, H].
// =====================================================================
__global__ __launch_bounds__(256) void
scan_emit(const float* __restrict__ A, const float* __restrict__ V,
          const float* __restrict__ P, float* __restrict__ out)
{
    int g     = blockIdx.x * 256 + threadIdx.x;     // B*NCHUNK*H threads
    int h     = g & (Hh - 1);
    int t     = g >> 9;
    int chunk = t & (NCHUNK - 1);
    int b     = t >> 6;

    size_t base  = ((size_t)(b * Tt + chunk * TC)) * Hh + h;
    size_t obase = ((size_t)(b * (Tt + 1) + chunk * TC + 1)) * Hh + h;
    float run = P[((size_t)(b * NCHUNK + chunk)) * Hh + h];

    if (chunk == 0)
        out[((size_t)b * (Tt + 1)) * Hh + h] = 0.5f;

    for (int i = 0; i < TC; ++i) {
        size_t p = base + (size_t)i * Hh;
        float a = A[p];
        run = lae_(run, V[p] - a);
        out[obase + (size_t)i * Hh] = __expf(a + run);
    }
}

// =====================================================================
extern "C" void kernel_launch(void* const* d_in, const int* in_sizes, int n_in,
                              void* d_out, int out_size, void* d_ws, size_t ws_size,
                              hipStream_t stream)
{
    const float* x   = (const float*)d_in[0];
    const float* Wf  = (const float*)d_in[1];
    const float* bfp = (const float*)d_in[2];
    const float* Wi  = (const float*)d_in[3];
    const float* bip = (const float*)d_in[4];
    const float* Wh  = (const float*)d_in[5];
    const float* bhp = (const float*)d_in[6];
    float* out = (float*)d_out;

    float* ws = (float*)d_ws;
    const size_t NA = (size_t)MTOT * Hh;            // 16M floats
    float* Aw = ws;                                 // log_f
    float* Vw = ws + NA;                            // log_i + log_g(zh)
    float* Sw = ws + 2 * NA;                        // chunk totals  [B,NCHUNK,H]
    float* Pw = Sw + (size_t)Bb * NCHUNK * Hh;      // chunk prefixes

    dim3 grid(MTOT / BM, Hh / BN);                  // 256 x 8 blocks
    minlstm_gemm<<<grid, 256, 0, stream>>>(x, Wf, bfp, Wi, bip, Wh, bhp, Aw, Vw);

    scan_chunk_total<<<(Bb * NCHUNK * Hh) / 256, 256, 0, stream>>>(Aw, Vw, Sw);
    scan_prefix     <<<(Bb * Hh) / 256,          256, 0, stream>>>(Sw, Pw);
    scan_emit       <<<(Bb * NCHUNK * Hh) / 256, 256, 0, stream>>>(Aw, Vw, Pw, out);
}